// SlicedTransport_87016037417479
// MI455X (gfx1250) — compile-verified
//
#include <hip/hip_runtime.h>

// MI455X / gfx1250 implementation of the sliced-transport reference.
//
//   data0 = data @ wT                      (fp32 WMMA GEMM #1 -> ws)
//   y0, logj = rq_spline(data0)            (elementwise, diff = y0-data0 in ws)
//   out = data + (y0 - data0) @ wT^T       (fp32 WMMA GEMM #2 -> d_out)
//   d_out tail = row-sums of logj
//
// Two GEMMs instead of the reference's three (algebraic fusion). Tiles are
// double-buffered in LDS and filled with GLOBAL_LOAD_ASYNC_TO_LDS_B128 (the
// CDNA5 async copy path, ASYNCcnt-tracked, issued via inline asm) so the DMA
// overlaps the WMMAs. Epilogue repacks accumulators through LDS for coalesced
// b128 stores.

typedef float v2f __attribute__((ext_vector_type(2)));
typedef float v8f __attribute__((ext_vector_type(8)));

#define USE_ASYNC 1   // inline-asm global_load_async_to_lds path

#define DD 512      // inner dim / n_components
#define KN 200      // spline knots per dim
#define BM 128      // block tile M
#define BN 64       // block tile N
#define KC 16       // K chunk staged in LDS
#define ASTRIDE 20  // padded LDS stride for A tile (conflict-free, 16B-alignable)
#define BSTRIDE 72  // padded LDS stride for B tile, [k][n] layout (TRANSB=0)
#define BTSTRIDE 20 // padded LDS stride for B tile, [n][k] layout (TRANSB=1)
#define CSTRIDE 68  // padded LDS stride for epilogue repack (8*68 % 64 == 32)
#define NIT (DD / KC)

#if USE_ASYNC
// Async DMA: 16B per lane, global -> LDS. VDST operand is the wave-relative
// LDS byte offset = low 32 bits of the flat shared-memory pointer (the LDS
// aperture truncates flat addresses to addr[31:0]).
__device__ __forceinline__ void async_b128(const float* g, float* l) {
    const unsigned lds_off = (unsigned)(uintptr_t)l;
    asm volatile("global_load_async_to_lds_b128 %0, %1, off"
                 :: "v"(lds_off), "v"(g) : "memory");
}
__device__ __forceinline__ void wait_async0() {
    asm volatile("s_wait_asynccnt 0x0" ::: "memory");
}
#endif

// ---------------------------------------------------------------------------
// Tiled fp32 GEMM using V_WMMA_F32_16X16X4_F32.
// C[M,512] = A[M,512] * B (+ addend), B = wT (TRANSB=0) or wT^T (TRANSB=1).
// 256 threads = 8 wave32s; wave grid 4x2, each wave owns a 32x32 patch
// (2x2 accumulators of 16x16). Double-buffered LDS tiles, async-copy filled.
// ---------------------------------------------------------------------------
template <bool TRANSB, bool ADDIN>
__global__ __launch_bounds__(256)
void gemm_wmma_f32(const float* __restrict__ A, const float* __restrict__ B,
                   const float* __restrict__ addend, float* __restrict__ C)
{
    constexpr int BSIZE = TRANSB ? (BN * BTSTRIDE) : (KC * BSTRIDE);
    // Tile phase uses <= 7680 floats; epilogue repack uses 128*68 = 8704.
    __shared__ float lds[BM * CSTRIDE];
    float* const As0 = lds;
    float* const As1 = lds + BM * ASTRIDE;
    float* const Bs0 = lds + 2 * BM * ASTRIDE;
    float* const Bs1 = Bs0 + BSIZE;

    const int t  = threadIdx.x;
    const int m0 = blockIdx.x * BM;
    const int n0 = blockIdx.y * BN;

    const int wid   = t >> 5;
    const int lane  = t & 31;
    const int lhalf = lane & 15;
    const int hi    = lane >> 4;        // 0: lanes 0-15, 1: lanes 16-31
    const int waveM = (wid & 3) * 32;   // 4 waves along M
    const int waveN = (wid >> 2) * 32;  // 2 waves along N

    // per-thread tile-copy coordinates (one b128 each; A gets two)
    const int arow = t >> 2;                              // 0..63 (+64)
    const int akq  = (t & 3) * 4;
    const int brow = TRANSB ? (t >> 2) : (t >> 4);        // [n] rows : [k] rows
    const int bcol = TRANSB ? ((t & 3) * 4) : ((t & 15) * 4);
    constexpr int bstr = TRANSB ? BTSTRIDE : BSTRIDE;

    auto gA0 = [&](int k0) { return A + (size_t)(m0 + arow) * DD + k0 + akq; };
    auto gA1 = [&](int k0) { return A + (size_t)(m0 + arow + 64) * DD + k0 + akq; };
    auto gBp = [&](int k0) {
        return TRANSB ? B + (size_t)(n0 + brow) * DD + k0 + bcol    // wT rows = out cols
                      : B + (size_t)(k0 + brow) * DD + n0 + bcol;
    };

    v8f acc[2][2] = {};

    auto computeChunk = [&](const float* Asc, const float* Bsc) {
        #pragma unroll
        for (int kk = 0; kk < KC; kk += 4) {
            const int ks = kk + hi * 2;   // lanes 16-31 hold K+2 per ISA layout
            v2f af[2], bf[2];
            #pragma unroll
            for (int tm = 0; tm < 2; ++tm) {
                const int row = waveM + tm * 16 + lhalf;
                af[tm].x = Asc[row * ASTRIDE + ks];
                af[tm].y = Asc[row * ASTRIDE + ks + 1];
            }
            #pragma unroll
            for (int tn = 0; tn < 2; ++tn) {
                const int col = waveN + tn * 16 + lhalf;
                if (TRANSB) {   // staged as-is [n][k]: adjacent pair -> ds_load_b64
                    bf[tn].x = Bsc[col * BTSTRIDE + ks];
                    bf[tn].y = Bsc[col * BTSTRIDE + ks + 1];
                } else {        // [k][n] layout
                    bf[tn].x = Bsc[ks * BSTRIDE + col];
                    bf[tn].y = Bsc[(ks + 1) * BSTRIDE + col];
                }
            }
            #pragma unroll
            for (int tm = 0; tm < 2; ++tm)
                #pragma unroll
                for (int tn = 0; tn < 2; ++tn)
                    acc[tm][tn] = __builtin_amdgcn_wmma_f32_16x16x4_f32(
                        false, af[tm], false, bf[tn],
                        (short)0, acc[tm][tn], false, false);
        }
    };

#if USE_ASYNC
    // ---- async-copy pipeline: DMA chunk it+1 while WMMAs chew chunk it ----
    auto issueTile = [&](int k0, float* Asb, float* Bsb) {
        async_b128(gA0(k0), Asb + arow * ASTRIDE + akq);
        async_b128(gA1(k0), Asb + (arow + 64) * ASTRIDE + akq);
        async_b128(gBp(k0), Bsb + brow * bstr + bcol);
    };
    issueTile(0, As0, Bs0);
    wait_async0();
    __syncthreads();
    for (int it = 0; it < NIT - 1; ++it) {
        float* Asc = (it & 1) ? As1 : As0;
        float* Bsc = (it & 1) ? Bs1 : Bs0;
        issueTile((it + 1) * KC, (it & 1) ? As0 : As1, (it & 1) ? Bs0 : Bs1);
        computeChunk(Asc, Bsc);          // 16 WMMAs per wave hide the async copy
        wait_async0();
        __syncthreads();
    }
    computeChunk(((NIT - 1) & 1) ? As1 : As0, ((NIT - 1) & 1) ? Bs1 : Bs0);
    __syncthreads();   // tiles dead; lds reused for repack below
#else
    // ---- fallback: register-staged double-buffered pipeline ----
    float4 va0, va1, vb;
    auto loadTile = [&](int k0) {
        va0 = *(const float4*)gA0(k0);
        va1 = *(const float4*)gA1(k0);
        vb  = *(const float4*)gBp(k0);
    };
    auto stageTile = [&](float* Asb, float* Bsb) {
        *(float4*)(Asb + arow * ASTRIDE + akq)        = va0;
        *(float4*)(Asb + (arow + 64) * ASTRIDE + akq) = va1;
        *(float4*)(Bsb + brow * bstr + bcol)          = vb;
    };
    loadTile(0);
    stageTile(As0, Bs0);
    __syncthreads();
    for (int it = 0; it < NIT - 1; ++it) {
        float* Asc = (it & 1) ? As1 : As0;
        float* Bsc = (it & 1) ? Bs1 : Bs0;
        loadTile((it + 1) * KC);
        computeChunk(Asc, Bsc);
        stageTile((it & 1) ? As0 : As1, (it & 1) ? Bs0 : Bs1);
        __syncthreads();
    }
    computeChunk(((NIT - 1) & 1) ? As1 : As0, ((NIT - 1) & 1) ? Bs1 : Bs0);
    __syncthreads();
#endif

    // ---- epilogue: repack through LDS for coalesced b128 stores ----
    #pragma unroll
    for (int tm = 0; tm < 2; ++tm) {
        #pragma unroll
        for (int tn = 0; tn < 2; ++tn) {
            const int col = waveN + tn * 16 + lhalf;
            #pragma unroll
            for (int r = 0; r < 8; ++r) {
                const int row = waveM + tm * 16 + r + hi * 8;   // C/D VGPR layout
                lds[row * CSTRIDE + col] = acc[tm][tn][r];
            }
        }
    }
    __syncthreads();
    #pragma unroll
    for (int j = 0; j < 8; ++j) {
        const int idx = t + j * 256;        // 0..2047 float4 slots
        const int row = idx >> 4;           // 0..127
        const int c4  = (idx & 15) * 4;     // 0..60
        float4 v = *(float4*)(lds + row * CSTRIDE + c4);
        const size_t g = (size_t)(m0 + row) * DD + n0 + c4;
        if (ADDIN) {
            const float4 ad = *(const float4*)(addend + g);
            v.x += ad.x; v.y += ad.y; v.z += ad.z; v.w += ad.w;
        }
        *(float4*)(C + g) = v;
    }
}

// ---------------------------------------------------------------------------
// Rational-quadratic spline: one row per block, 2 dims per thread.
// Reads data0 from ws, writes diff = y0 - data0 in place, reduces logj row-sum.
// ---------------------------------------------------------------------------
__global__ __launch_bounds__(256)
void rq_spline_kernel(const float* __restrict__ data0, float* __restrict__ diff,
                      const float* __restrict__ xx, const float* __restrict__ yy,
                      const float* __restrict__ delta, float* __restrict__ logjsum)
{
    __shared__ float red[256];
    const int n = blockIdx.x;
    const int t = threadIdx.x;
    float lsum = 0.f;

    #pragma unroll
    for (int i = 0; i < 2; ++i) {
        const int d = t + i * 256;
        const float x = data0[(size_t)n * DD + d];
        const float* xd = xx + (size_t)d * KN;
        const float* yd = yy + (size_t)d * KN;
        const float* dl = delta + (size_t)d * KN;

        // searchsorted side='left': first index with xd[idx] >= x
        int lo = 0, hib = KN;
        while (lo < hib) {
            const int mid = (lo + hib) >> 1;
            if (xd[mid] < x) lo = mid + 1; else hib = mid;
        }
        const int idx = lo;

        float y, ld;
        if (idx == 0) {                       // left linear extrapolation
            y  = yd[0] + (x - xd[0]) * dl[0];
            ld = __logf(dl[0]);
        } else if (idx == KN) {               // right linear extrapolation
            y  = yd[KN - 1] + (x - xd[KN - 1]) * dl[KN - 1];
            ld = __logf(dl[KN - 1]);
        } else {                              // interior RQ segment
            int k = idx - 1;
            if (k > KN - 2) k = KN - 2;
            const float xk = xd[k], xk1 = xd[k + 1];
            const float yk = yd[k], yk1 = yd[k + 1];
            const float dk = dl[k], dk1 = dl[k + 1];
            const float w  = xk1 - xk;
            const float s  = (yk1 - yk) / w;
            float xi = (x - xk) / w;
            xi = fminf(fmaxf(xi, 0.f), 1.f);
            const float om  = 1.f - xi;
            const float den = s + (dk1 + dk - 2.f * s) * xi * om;
            y = yk + (yk1 - yk) * (s * xi * xi + dk * xi * om) / den;
            const float dv = s * s * (dk1 * xi * xi + 2.f * s * xi * om + dk * om * om)
                             / (den * den);
            ld = __logf(dv);
        }
        diff[(size_t)n * DD + d] = y - x;     // y0 - data0, in place
        lsum += ld;
    }

    red[t] = lsum;
    __syncthreads();
    #pragma unroll
    for (int s = 128; s > 0; s >>= 1) {
        if (t < s) red[t] += red[t + s];
        __syncthreads();
    }
    if (t == 0) logjsum[n] = red[0];
}

// ---------------------------------------------------------------------------
extern "C" void kernel_launch(void* const* d_in, const int* in_sizes, int n_in,
                              void* d_out, int out_size, void* d_ws, size_t ws_size,
                              hipStream_t stream)
{
    (void)n_in; (void)out_size; (void)ws_size;
    const float* data  = (const float*)d_in[0];   // [N, 512]
    const float* wT    = (const float*)d_in[1];   // [512, 512]
    const float* xx    = (const float*)d_in[2];   // [512, 200]
    const float* yy    = (const float*)d_in[3];   // [512, 200]
    const float* delta = (const float*)d_in[4];   // [512, 200]
    float* out  = (float*)d_out;                  // [N*512] out, then [N] logj sums
    float* ws   = (float*)d_ws;                   // N*512 floats: data0, then diff

    const int N = in_sizes[0] / DD;               // 65536
    dim3 grid(N / BM, DD / BN);                   // (512, 8)

    // 1) data0 = data @ wT  -> ws
    gemm_wmma_f32<false, false><<<grid, 256, 0, stream>>>(data, wT, nullptr, ws);
    // 2) spline: ws -> diff (in place), logj row-sums -> out[N*512 ..)
    rq_spline_kernel<<<N, 256, 0, stream>>>(ws, ws, xx, yy, delta, out + (size_t)N * DD);
    // 3) out = data + diff @ wT^T
    gemm_wmma_f32<true, true><<<grid, 256, 0, stream>>>(ws, wT, data, out);
}